// Attention_48163763257567
// MI455X (gfx1250) — compile-verified
//
#include <hip/hip_runtime.h>
#include <math.h>

typedef __attribute__((ext_vector_type(2))) float v2f;
typedef __attribute__((ext_vector_type(8))) float v8f;

// Problem sizes (match reference)
constexpr int T = 32768, B = 1024, E = 64, H = 36;
constexpr int KP = 48;                 // H padded to 3 N-tiles of 16
constexpr int NP = E * KP + KP;        // 3120 padded cols of Gc = [G | c]
constexpr int NT1 = NP / 16;           // 195 N tiles for build-G GEMM
constexpr int MT1 = B / 16;            // 64  M tiles

// Workspace layout (bytes); total ~17.7 MB
constexpr size_t OFF_GC   = 0;                               // B*NP f32
constexpr size_t OFF_H    = OFF_GC + (size_t)B * NP * 4;     // T*H f32
constexpr size_t OFF_STAT = OFF_H + (size_t)T * H * 4;       // avg@+0, invstd@+256B
constexpr size_t OFF_CNT  = OFF_STAT + 1024;                 // B int
constexpr size_t OFF_CUR  = OFF_CNT + 4096;                  // B int (contiguous w/ CNT)
constexpr size_t OFF_OFFS = OFF_CUR + 4096;                  // B+1 int
constexpr size_t OFF_SORT = OFF_OFFS + 8192;                 // T int

__global__ void kZero(int* __restrict__ p, int n) {
  int i = blockIdx.x * 256 + threadIdx.x;
  if (i < n) p[i] = 0;
}

__global__ void kHist(const int* __restrict__ tok, int* __restrict__ cnt) {
  int t = blockIdx.x * 256 + threadIdx.x;
  if (t < T) atomicAdd(&cnt[tok[t]], 1);
}

// Single-block Hillis-Steele inclusive scan over 1024 bucket counts.
__global__ void __launch_bounds__(1024) kScan(const int* __restrict__ cnt,
                                              int* __restrict__ offs) {
  __shared__ int sc[1024];
  int tid = threadIdx.x;
  sc[tid] = cnt[tid];
  __syncthreads();
  for (int off = 1; off < 1024; off <<= 1) {
    int add = (tid >= off) ? sc[tid - off] : 0;
    __syncthreads();
    sc[tid] += add;
    __syncthreads();
  }
  offs[tid + 1] = sc[tid];
  if (tid == 0) offs[0] = 0;
}

__global__ void kScatter(const int* __restrict__ tok, const int* __restrict__ offs,
                         int* __restrict__ cur, int* __restrict__ sorted) {
  int t = blockIdx.x * 256 + threadIdx.x;
  if (t < T) {
    int b = tok[t];
    int p = offs[b] + atomicAdd(&cur[b], 1);
    sorted[p] = t;
  }
}

// GEMM 1: Gc[b, n] = sum_j query[b,j] * Bmat[j,n] + bias(n)
//   n = i*48+k (i<64): Bmat[j,n] = W1[(64 + i*64 + j)*36 + k], bias = W1[i*36+k]
//   n = 3072+k:        Bmat[j,n] = W1[(4160 + j)*36 + k],      bias = b1[k]
// One wave per 16x16 tile; K=64 via 16 x V_WMMA_F32_16X16X4_F32.
__global__ void __launch_bounds__(32) kBuildGc(const float* __restrict__ query,
                                               const float* __restrict__ W1,
                                               const float* __restrict__ b1,
                                               float* __restrict__ Gc) {
  const int lane = threadIdx.x;          // 0..31 (wave32)
  const int lm = lane & 15, half = lane >> 4;
  const int n  = blockIdx.x * 16 + lm;   // padded output column
  const int m0 = blockIdx.y * 16;        // query-row tile base
  const int i  = n / KP;                 // 0..63 = G region, 64 = c region
  const int k  = n % KP;
  const bool kv = (k < H);
  const float msk = kv ? 1.f : 0.f;
  const int kk = kv ? k : 0;             // clamped (keeps loads in-range)
  const int rowStart = (i < E) ? (E + i * E) : (E + E * E);
  const float* bsrc = W1 + (size_t)rowStart * H + kk;
  const float bias = msk * ((i < E) ? W1[i * H + kk] : b1[kk]);

  // B fragments for all 16 K-steps: lanes 0-15 hold K={4s,4s+1}, 16-31 hold K={4s+2,4s+3}
  v2f barr[16];
#pragma unroll
  for (int s = 0; s < 16; ++s) {
    const int j0 = 4 * s + 2 * half;
    v2f bv;
    bv.x = msk * bsrc[(size_t)j0 * H];
    bv.y = msk * bsrc[(size_t)(j0 + 1) * H];
    barr[s] = bv;
  }
  // A fragments: row = query row (m0+lm), same K split across lane halves
  const float* qrow = query + (size_t)(m0 + lm) * E;
  v2f aarr[16];
#pragma unroll
  for (int s = 0; s < 16; ++s) {
    const int j0 = 4 * s + 2 * half;
    v2f av; av.x = qrow[j0]; av.y = qrow[j0 + 1];
    aarr[s] = av;
  }
  v8f c;
#pragma unroll
  for (int r = 0; r < 8; ++r) c[r] = bias;   // bias constant across M rows
#pragma unroll
  for (int s = 0; s < 16; ++s)
    c = __builtin_amdgcn_wmma_f32_16x16x4_f32(false, aarr[s], false, barr[s],
                                              (short)0, c, false, false);
#pragma unroll
  for (int r = 0; r < 8; ++r)
    Gc[(size_t)(m0 + r + 8 * half) * NP + n] = c[r];
}

// GEMM 2: per bucket b, h[tok, k] = x[tok,:] @ G[b] + c[b,k].
// Block = 3 waves; wave w owns N-tile w (cols 16w..16w+15), loops bucket M-tiles.
// B fragments (G[b]) loaded once per wave and reused across all M tiles.
__global__ void __launch_bounds__(96) kBucketGemm(const float* __restrict__ x,
                                                  const float* __restrict__ Gc,
                                                  const int* __restrict__ offs,
                                                  const int* __restrict__ sorted,
                                                  float* __restrict__ h) {
  const int b = blockIdx.x;
  const int lane = threadIdx.x & 31, ntile = threadIdx.x >> 5;  // ntile 0..2
  const int lm = lane & 15, half = lane >> 4;
  const int start = offs[b];
  const int count = offs[b + 1] - start;
  if (count == 0) return;                 // uniform across block -> EXEC stays full
  const int n = ntile * 16 + lm;          // output col 0..47 (k padded)
  const float* gb = Gc + (size_t)b * NP;

  v2f barr[16];
#pragma unroll
  for (int s = 0; s < 16; ++s) {
    const int i0 = 4 * s + 2 * half;      // K dim = i (x-feature)
    v2f bv; bv.x = gb[i0 * KP + n]; bv.y = gb[(i0 + 1) * KP + n];
    barr[s] = bv;
  }
  const float cval = gb[E * KP + n];      // c[b,k] (zero in padding)

  const int mtiles = (count + 15) >> 4;
  for (int mt = 0; mt < mtiles; ++mt) {
    const int row = mt * 16 + lm;
    const int slot = start + ((row < count) ? row : (count - 1));  // branch-free clamp
    const int tok = sorted[slot];
    const float* xr = x + (size_t)tok * E;
    v8f c;
#pragma unroll
    for (int r = 0; r < 8; ++r) c[r] = cval;
#pragma unroll
    for (int s = 0; s < 16; ++s) {
      const int j0 = 4 * s + 2 * half;
      v2f av; av.x = xr[j0]; av.y = xr[j0 + 1];
      c = __builtin_amdgcn_wmma_f32_16x16x4_f32(false, av, false, barr[s],
                                                (short)0, c, false, false);
    }
    if (n < H) {
#pragma unroll
      for (int r = 0; r < 8; ++r) {
        const int rr = mt * 16 + r + 8 * half;   // D layout: M = r + 8*half
        if (rr < count) {
          const int tt = sorted[start + rr];
          h[(size_t)tt * H + n] = c[r];
        }
      }
    }
  }
}

// Deterministic per-column mean / unbiased-invstd over all T tokens.
__global__ void __launch_bounds__(256) kStats(const float* __restrict__ h,
                                              float* __restrict__ stat) {
  const int c = blockIdx.x;   // 0..35
  const int tid = threadIdx.x;
  float s1 = 0.f, s2 = 0.f;
  for (int t = tid; t < T; t += 256) {        // fixed traversal order
    float v = h[(size_t)t * H + c];
    s1 += v; s2 += v * v;
  }
  __shared__ float a1[256], a2[256];
  a1[tid] = s1; a2[tid] = s2;
  __syncthreads();
  for (int off = 128; off > 0; off >>= 1) {   // fixed tree order
    if (tid < off) { a1[tid] += a1[tid + off]; a2[tid] += a2[tid + off]; }
    __syncthreads();
  }
  if (tid == 0) {
    float sum = a1[0], ssq = a2[0];
    float avg = sum / (float)T;
    float var = (ssq - sum * avg) / (float)(T - 1);  // ddof=1
    stat[c] = avg;
    stat[64 + c] = rsqrtf(var);
  }
}

// Dice + final projection: out[t] = sum_k dice(h)[t,k] * W2[k] + b2
__global__ void __launch_bounds__(256) kOut(const float* __restrict__ h,
                                            const float* __restrict__ stat,
                                            const float* __restrict__ W2,
                                            const float* __restrict__ b2,
                                            const float* __restrict__ alpha,
                                            float* __restrict__ out) {
  int t = blockIdx.x * 256 + threadIdx.x;
  if (t >= T) return;
  const float a = alpha[0];
  float acc = b2[0];
  const float* hr = h + (size_t)t * H;
#pragma unroll
  for (int k = 0; k < H; ++k) {
    float hv = hr[k];
    float z = (hv - stat[k]) * stat[64 + k];
    float p = 1.f / (1.f + __expf(-z));
    acc += (hv * p + a * hv * (1.f - p)) * W2[k];
  }
  out[t] = acc;
}

extern "C" void kernel_launch(void* const* d_in, const int* in_sizes, int n_in,
                              void* d_out, int out_size, void* d_ws, size_t ws_size,
                              hipStream_t stream) {
  const float* x     = (const float*)d_in[0];
  const float* query = (const float*)d_in[1];
  const int*   tok   = (const int*)d_in[2];
  const float* W1    = (const float*)d_in[3];
  const float* b1    = (const float*)d_in[4];
  const float* alpha = (const float*)d_in[5];
  const float* W2    = (const float*)d_in[6];
  const float* b2    = (const float*)d_in[7];
  float* out = (float*)d_out;

  char* ws = (char*)d_ws;
  float* Gc     = (float*)(ws + OFF_GC);
  float* hbuf   = (float*)(ws + OFF_H);
  float* stat   = (float*)(ws + OFF_STAT);
  int*   cnt    = (int*)(ws + OFF_CNT);
  int*   cur    = (int*)(ws + OFF_CUR);
  int*   offs   = (int*)(ws + OFF_OFFS);
  int*   sorted = (int*)(ws + OFF_SORT);

  (void)in_sizes; (void)n_in; (void)out_size; (void)ws_size;

  // 1) counting sort of tokens by query index
  kZero<<<8, 256, 0, stream>>>(cnt, 2048);                 // cnt + cur (contiguous)
  kHist<<<(T + 255) / 256, 256, 0, stream>>>(tok, cnt);
  kScan<<<1, 1024, 0, stream>>>(cnt, offs);
  kScatter<<<(T + 255) / 256, 256, 0, stream>>>(tok, offs, cur, sorted);
  // 2) WMMA GEMM: fold W1 against every query row -> Gc
  kBuildGc<<<dim3(NT1, MT1), 32, 0, stream>>>(query, W1, b1, Gc);
  // 3) WMMA GEMM per bucket: h = x @ G[b] + c[b]
  kBucketGemm<<<B, 96, 0, stream>>>(x, Gc, offs, sorted, hbuf);
  // 4) Dice statistics (deterministic) + output projection
  kStats<<<H, 256, 0, stream>>>(hbuf, stat);
  kOut<<<(T + 255) / 256, 256, 0, stream>>>(hbuf, stat, W2, b2, alpha, out);
}